// EdgeGNNLayer_44006234914855
// MI455X (gfx1250) — compile-verified
//
#include <hip/hip_runtime.h>

// ---------------------------------------------------------------------------
// EdgeGNN layer for MI455X (gfx1250, wave32, WMMA)
//   agg  = segment_mean(node_feat[src], dst)           (scatter phase)
//   h    = relu(relu([nf|agg] @ W1 + b1) @ W2 + b2)    (fp32 WMMA phase)
//
// MLP tiling: 64 rows/block, 8 waves; wave w owns N-tile [16w,16w+16) and
// computes FOUR 16x16 M-tiles against one shared B fragment -> 4x less L2
// weight traffic (~150 MB vs ~600 MB) and 4 independent WMMA chains.
// Uses ~98 KB LDS, which only CDNA5's 320 KB/WGP can host.
// ---------------------------------------------------------------------------

typedef __attribute__((ext_vector_type(2))) float v2f;
typedef __attribute__((ext_vector_type(8))) float v8f;

#define N_NODES 50000
#define N_EDGES 800000
#define D       128
#define D2      256   // 2*D
#define BR      64    // rows per block
#define XST     (D2 + 4)   // Xs row stride (floats): 16B-aligned, bank step 4
#define HST     (D + 4)    // Hs row stride

// ---------------------------------------------------------------------------
__global__ void zero_kernel(float* __restrict__ p, int n) {
    int i = blockIdx.x * blockDim.x + threadIdx.x;
    if (i < n) p[i] = 0.0f;
}

// ---------------------------------------------------------------------------
// Scatter: one wave32 per edge; float4 gather + 4x global_atomic_add_f32.
// node_feat and agg are both L2-resident (25.6 MB each on 192 MB L2).
// ---------------------------------------------------------------------------
__global__ __launch_bounds__(256)
void scatter_kernel(const float* __restrict__ nf,
                    const int*   __restrict__ src,
                    const int*   __restrict__ dst,
                    float* __restrict__ agg,
                    float* __restrict__ deg) {
    int edge = (blockIdx.x * blockDim.x + threadIdx.x) >> 5;   // wave32
    int lane = threadIdx.x & 31;
    if (edge >= N_EDGES) return;
    int s = src[edge];
    int d = dst[edge];
    const float4 v = *(const float4*)(nf + (size_t)s * D + lane * 4);
    float* a = agg + (size_t)d * D + lane * 4;
    atomicAdd(a + 0, v.x);
    atomicAdd(a + 1, v.y);
    atomicAdd(a + 2, v.z);
    atomicAdd(a + 3, v.w);
    if (lane == 0) atomicAdd(deg + d, 1.0f);
}

// ---------------------------------------------------------------------------
// Fused MLP with V_WMMA_F32_16X16X4_F32.
// A (16x4 f32): lane m=(lane&15), kb=(lane>>4)*2 -> {A[m][k+kb], A[m][k+kb+1]}
// B (4x16 f32): lane n=(lane&15), same kb        -> {B[k+kb][n], B[k+kb+1][n]}
// C/D (16x16 f32): vgpr r -> M = r + 8*(lane>>4), N = lane&15
// ---------------------------------------------------------------------------
__global__ __launch_bounds__(256)
void mlp_kernel(const float* __restrict__ nf,
                const float* __restrict__ agg,
                const float* __restrict__ deg,
                const float* __restrict__ W1, const float* __restrict__ b1,
                const float* __restrict__ W2, const float* __restrict__ b2,
                float* __restrict__ out) {
    __shared__ float Xs[BR][XST];   // 64 x 260 x 4B = 65 KB
    __shared__ float Hs[BR][HST];   // 64 x 132 x 4B = 33 KB
    __shared__ float invd[BR];

    const int tid  = threadIdx.x;
    const int wave = tid >> 5;
    const int lane = tid & 31;
    const int row0 = blockIdx.x * BR;

    // ---- per-row 1/max(deg,1) --------------------------------------------
    for (int r = tid; r < BR; r += 256) {
        int row = row0 + r;
        float dg = (row < N_NODES) ? deg[row] : 1.0f;
        invd[r] = 1.0f / fmaxf(dg, 1.0f);
    }
    __syncthreads();

    // ---- stage X = [node_feat | agg*invd] (float4, 16 slots/thread) ------
    for (int e = tid; e < BR * (D2 / 4); e += 256) {
        int r  = e >> 6;          // 64 float4-slots per row
        int c  = (e & 63) * 4;
        int row = row0 + r;
        float4 v = make_float4(0.f, 0.f, 0.f, 0.f);
        if (row < N_NODES) {
            if (c < D) {
                v = *(const float4*)(nf + (size_t)row * D + c);
            } else {
                float4 t = *(const float4*)(agg + (size_t)row * D + (c - D));
                float iv = invd[r];
                v = make_float4(t.x * iv, t.y * iv, t.z * iv, t.w * iv);
            }
        }
        *(float4*)&Xs[r][c] = v;
    }
    __syncthreads();

    const int n0 = wave * 16;
    const int mA = lane & 15;
    const int kb = (lane >> 4) * 2;
    const int nB = lane & 15;

    // ---- GEMM1: H = relu(X @ W1 + b1), K = 256, 4 M-tiles per wave --------
    v8f c0 = {}, c1 = {}, c2 = {}, c3 = {};
    for (int k0 = 0; k0 < D2; k0 += 4) {
        v2f b;
        b.x = W1[(size_t)(k0 + kb)     * D + n0 + nB];
        b.y = W1[(size_t)(k0 + kb + 1) * D + n0 + nB];
        v2f a0, a1, a2, a3;
        a0.x = Xs[mA +  0][k0 + kb]; a0.y = Xs[mA +  0][k0 + kb + 1];
        a1.x = Xs[mA + 16][k0 + kb]; a1.y = Xs[mA + 16][k0 + kb + 1];
        a2.x = Xs[mA + 32][k0 + kb]; a2.y = Xs[mA + 32][k0 + kb + 1];
        a3.x = Xs[mA + 48][k0 + kb]; a3.y = Xs[mA + 48][k0 + kb + 1];
        c0 = __builtin_amdgcn_wmma_f32_16x16x4_f32(false, a0, false, b, (short)0, c0, false, false);
        c1 = __builtin_amdgcn_wmma_f32_16x16x4_f32(false, a1, false, b, (short)0, c1, false, false);
        c2 = __builtin_amdgcn_wmma_f32_16x16x4_f32(false, a2, false, b, (short)0, c2, false, false);
        c3 = __builtin_amdgcn_wmma_f32_16x16x4_f32(false, a3, false, b, (short)0, c3, false, false);
    }
    {
        float bias  = b1[n0 + nB];
        int   mbase = (lane >> 4) * 8;
#pragma unroll
        for (int r = 0; r < 8; ++r) {
            Hs[ 0 + mbase + r][n0 + nB] = fmaxf(c0[r] + bias, 0.0f);
            Hs[16 + mbase + r][n0 + nB] = fmaxf(c1[r] + bias, 0.0f);
            Hs[32 + mbase + r][n0 + nB] = fmaxf(c2[r] + bias, 0.0f);
            Hs[48 + mbase + r][n0 + nB] = fmaxf(c3[r] + bias, 0.0f);
        }
    }
    __syncthreads();

    // ---- GEMM2: out = relu(H @ W2 + b2), K = 128 --------------------------
    v8f d0 = {}, d1 = {}, d2 = {}, d3 = {};
    for (int k0 = 0; k0 < D; k0 += 4) {
        v2f b;
        b.x = W2[(size_t)(k0 + kb)     * D + n0 + nB];
        b.y = W2[(size_t)(k0 + kb + 1) * D + n0 + nB];
        v2f a0, a1, a2, a3;
        a0.x = Hs[mA +  0][k0 + kb]; a0.y = Hs[mA +  0][k0 + kb + 1];
        a1.x = Hs[mA + 16][k0 + kb]; a1.y = Hs[mA + 16][k0 + kb + 1];
        a2.x = Hs[mA + 32][k0 + kb]; a2.y = Hs[mA + 32][k0 + kb + 1];
        a3.x = Hs[mA + 48][k0 + kb]; a3.y = Hs[mA + 48][k0 + kb + 1];
        d0 = __builtin_amdgcn_wmma_f32_16x16x4_f32(false, a0, false, b, (short)0, d0, false, false);
        d1 = __builtin_amdgcn_wmma_f32_16x16x4_f32(false, a1, false, b, (short)0, d1, false, false);
        d2 = __builtin_amdgcn_wmma_f32_16x16x4_f32(false, a2, false, b, (short)0, d2, false, false);
        d3 = __builtin_amdgcn_wmma_f32_16x16x4_f32(false, a3, false, b, (short)0, d3, false, false);
    }
    {
        float bias  = b2[n0 + nB];
        int   mbase = (lane >> 4) * 8;
#pragma unroll
        for (int t = 0; t < 4; ++t) {
            v8f acc = (t == 0) ? d0 : (t == 1) ? d1 : (t == 2) ? d2 : d3;
#pragma unroll
            for (int r = 0; r < 8; ++r) {
                int row = row0 + t * 16 + mbase + r;
                if (row < N_NODES) {
                    out[(size_t)row * D + n0 + nB] = fmaxf(acc[r] + bias, 0.0f);
                }
            }
        }
    }
}

// ---------------------------------------------------------------------------
extern "C" void kernel_launch(void* const* d_in, const int* in_sizes, int n_in,
                              void* d_out, int out_size, void* d_ws, size_t ws_size,
                              hipStream_t stream) {
    const float* nf = (const float*)d_in[0];
    const int*   ei = (const int*)d_in[1];
    const float* W1 = (const float*)d_in[2];
    const float* b1 = (const float*)d_in[3];
    const float* W2 = (const float*)d_in[4];
    const float* b2 = (const float*)d_in[5];
    float*       out = (float*)d_out;

    float* agg = (float*)d_ws;                    // [N_NODES * D]
    float* deg = agg + (size_t)N_NODES * D;       // [N_NODES]

    const int zn      = N_NODES * D + N_NODES;
    zero_kernel<<<(zn + 255) / 256, 256, 0, stream>>>(agg, zn);

    const int sblocks = (N_EDGES * 32 + 255) / 256;   // one wave32 per edge
    scatter_kernel<<<sblocks, 256, 0, stream>>>(nf, ei, ei + N_EDGES, agg, deg);

    const int mblocks = (N_NODES + BR - 1) / BR;      // 782
    mlp_kernel<<<mblocks, 256, 0, stream>>>(nf, agg, deg, W1, b1, W2, b2, out);
}